// MultiScaleAdaptiveElasticityLossWithLame_29703993819925
// MI455X (gfx1250) — compile-verified
//
#include <hip/hip_runtime.h>
#include <stdint.h>

// ---------------------------------------------------------------------------
// Multi-scale adaptive elasticity loss, MI455X (gfx1250).
// Bandwidth-bound stencil; uses CDNA5 async global->LDS copies + s_wait_asynccnt.
// ---------------------------------------------------------------------------

#define TX 8
#define TY 8
#define TZ 32
#define LX (TX + 2)      // 10
#define LY (TY + 2)      // 10
#define LZ (TZ + 2)      // 34 used
#define LZP 36           // padded z stride (bank-conflict avoidance)
#define NCH 4            // u, v, w, img
#define LDS_FLOATS (NCH * LX * LY * LZP)   // 14400 floats = 57.6 KB

typedef __attribute__((address_space(1))) int gint_t;
typedef __attribute__((address_space(3))) int lint_t;

__device__ __forceinline__ void async_g2l_b32(const float* gp, const float* lp) {
#if __has_builtin(__builtin_amdgcn_global_load_async_to_lds_b32)
  // (global src, lds dst, imm offset, imm cpol)
  __builtin_amdgcn_global_load_async_to_lds_b32(
      reinterpret_cast<gint_t*>(reinterpret_cast<uintptr_t>(gp)),
      reinterpret_cast<lint_t*>(
          static_cast<uint32_t>(reinterpret_cast<uintptr_t>(lp))),
      0, 0);
#else
  uint32_t loff = (uint32_t)(uintptr_t)lp;   // low 32 bits of flat LDS addr == LDS offset
  asm volatile("global_load_async_to_lds_b32 %0, %1, off"
               :: "v"(loff), "v"(gp) : "memory");
#endif
}

__device__ __forceinline__ void wait_async_zero() {
#if __has_builtin(__builtin_amdgcn_s_wait_asynccnt)
  __builtin_amdgcn_s_wait_asynccnt(0);
#else
  asm volatile("s_wait_asynccnt 0x0" ::: "memory");
#endif
}

__device__ __forceinline__ int clampi(int v, int lo, int hi) {
  return v < lo ? lo : (v > hi ? hi : v);
}

// ---------------------------------------------------------------------------
// Tiled stencil + per-block partial sum.
// def layout: [B][3][D0][D1][D2], img layout: [B][1][D0][D1][D2]
// ---------------------------------------------------------------------------
__global__ __launch_bounds__(256) void elasticity_tile_kernel(
    const float* __restrict__ def, const float* __restrict__ img,
    int D0, int D1, int D2, int tilesY, int tilesZ, int ntiles,
    float* __restrict__ psum)
{
  __shared__ float lds[LDS_FLOATS];

  const int tid = threadIdx.x;
  const int b = blockIdx.y;
  int tile = blockIdx.x;
  const int tz = tile % tilesZ; tile /= tilesZ;
  const int ty = tile % tilesY; tile /= tilesY;
  const int tx = tile;
  const int x0 = tx * TX, y0 = ty * TY, z0 = tz * TZ;

  const long long vol = (long long)D0 * D1 * D2;

  // -------- stage haloed tile (clamped) into LDS with async copies --------
  const int nLoad = NCH * LX * LY * LZ;   // 13600
  for (int t = tid; t < nLoad; t += 256) {
    int zz = t % LZ;
    int r = t / LZ;
    int yy = r % LY; r /= LY;
    int xx = r % LX;
    int ch = r / LX;
    int gx = clampi(x0 - 1 + xx, 0, D0 - 1);
    int gy = clampi(y0 - 1 + yy, 0, D1 - 1);
    int gz = clampi(z0 - 1 + zz, 0, D2 - 1);
    const float* srcp = (ch == 3) ? (img + (long long)b * vol)
                                  : (def + ((long long)b * 3 + ch) * vol);
    const float* g = srcp + ((long long)gx * D1 + gy) * D2 + gz;
    const float* l = &lds[((ch * LX + xx) * LY + yy) * LZP + zz];
    async_g2l_b32(g, l);
  }
  wait_async_zero();
  __syncthreads();

  // -------- compute: each thread owns fixed (ly,lz), loops over lx --------
  const int lz = tid % TZ;       // 0..31
  const int ly = tid / TZ;       // 0..7
  const int yy = ly + 1, zz = lz + 1;
  const int gy = y0 + ly, gz = z0 + lz;
  const bool yzok = (gy < D1) && (gz < D2);
  const float wy = (gy == 0 || gy == D1 - 1) ? 1.0f : 0.5f;
  const float wz = (gz == 0 || gz == D2 - 1) ? 1.0f : 0.5f;

  float acc = 0.0f;
  if (yzok) {
    for (int lx = 0; lx < TX; ++lx) {
      int gx = x0 + lx;
      if (gx >= D0) break;
      float wx = (gx == 0 || gx == D0 - 1) ? 1.0f : 0.5f;
      int xx = lx + 1;

      float dX[NCH], dY[NCH], dZ[NCH];
#pragma unroll
      for (int ch = 0; ch < NCH; ++ch) {
        int c = ((ch * LX + xx) * LY + yy) * LZP + zz;
        float xm = lds[c - LY * LZP], xp = lds[c + LY * LZP];
        float ym = lds[c - LZP],      yp = lds[c + LZP];
        float zm = lds[c - 1],        zp = lds[c + 1];
        // clamped halo => (xp - xm) already equals the one-sided diff at edges
        dX[ch] = (xp - xm) * wx;
        dY[ch] = (yp - ym) * wy;
        dZ[ch] = (zp - zm) * wz;
      }

      float Exx = dX[0], Eyy = dY[1], Ezz = dZ[2];
      float Exy = 0.5f * (dY[0] + dX[1]);
      float Exz = 0.5f * (dZ[0] + dX[2]);
      float Eyz = 0.5f * (dZ[1] + dY[2]);
      float tr  = Exx + Eyy + Ezz;
      float gmag = sqrtf(dX[3] * dX[3] + dY[3] * dY[3] + dZ[3] * dZ[3]);
      float lam = 1.0f + 0.5f * gmag;   // LAMBDA_0 + KAPPA_LAMBDA * g
      float mu  = 1.0f + 0.5f * gmag;   // MU_0 + KAPPA_MU * g
      float energy = 0.5f * lam * tr * tr
                   + mu * (Exx * Exx + Eyy * Eyy + Ezz * Ezz
                           + 2.0f * (Exy * Exy + Exz * Exz + Eyz * Eyz));
      float w = 1.0f + 0.1f * gmag;     // BASE_WEIGHT + GRADIENT_SCALING * g
      acc += w * energy;
    }
  }

  // -------- deterministic block reduction (reuse LDS) --------
  __syncthreads();
  lds[tid] = acc;
  __syncthreads();
  for (int s = 128; s > 0; s >>= 1) {
    if (tid < s) lds[tid] += lds[tid + s];
    __syncthreads();
  }
  if (tid == 0) psum[(long long)b * ntiles + blockIdx.x] = lds[0];
}

// ---------------------------------------------------------------------------
// Align-corners trilinear downsample from ORIGINAL volumes (matches the
// reference's sequential per-axis resize1d, which is separable).
// Outputs: def_out [B][3][O0][O1][O2], img_out [B][1][O0][O1][O2]
// ---------------------------------------------------------------------------
__global__ __launch_bounds__(256) void downsample_kernel(
    const float* __restrict__ def, const float* __restrict__ img,
    int I0, int I1, int I2, int O0, int O1, int O2,
    float r0, float r1, float r2,
    float* __restrict__ def_out, float* __restrict__ img_out)
{
  const long long ivol = (long long)I0 * I1 * I2;
  const long long ovol = (long long)O0 * O1 * O2;
  const long long total = 2LL * NCH * ovol;   // B = 2
  for (long long i = (long long)blockIdx.x * blockDim.x + threadIdx.x;
       i < total; i += (long long)gridDim.x * blockDim.x) {
    long long idx = i;
    int z = (int)(idx % O2); idx /= O2;
    int y = (int)(idx % O1); idx /= O1;
    int x = (int)(idx % O0); idx /= O0;
    int ch = (int)(idx % NCH); idx /= NCH;
    int b = (int)idx;

    float c0 = (float)x * r0, c1 = (float)y * r1, c2 = (float)z * r2;
    int x0i = clampi((int)floorf(c0), 0, I0 - 1);
    int y0i = clampi((int)floorf(c1), 0, I1 - 1);
    int z0i = clampi((int)floorf(c2), 0, I2 - 1);
    int x1i = min(x0i + 1, I0 - 1);
    int y1i = min(y0i + 1, I1 - 1);
    int z1i = min(z0i + 1, I2 - 1);
    float fx = c0 - (float)x0i, fy = c1 - (float)y0i, fz = c2 - (float)z0i;

    const float* src = (ch == 3) ? (img + (long long)b * ivol)
                                 : (def + ((long long)b * 3 + ch) * ivol);
    long long p00 = ((long long)x0i * I1 + y0i) * I2;
    long long p01 = ((long long)x0i * I1 + y1i) * I2;
    long long p10 = ((long long)x1i * I1 + y0i) * I2;
    long long p11 = ((long long)x1i * I1 + y1i) * I2;

    float v000 = src[p00 + z0i], v001 = src[p00 + z1i];
    float v010 = src[p01 + z0i], v011 = src[p01 + z1i];
    float v100 = src[p10 + z0i], v101 = src[p10 + z1i];
    float v110 = src[p11 + z0i], v111 = src[p11 + z1i];

    float a0 = v000 + fz * (v001 - v000);
    float a1 = v010 + fz * (v011 - v010);
    float b0 = v100 + fz * (v101 - v100);
    float b1 = v110 + fz * (v111 - v110);
    float c  = a0 + fy * (a1 - a0);
    float d  = b0 + fy * (b1 - b0);
    float val = c + fx * (d - c);

    long long o3 = ((long long)x * O1 + y) * O2 + z;
    if (ch < 3) def_out[((long long)b * 3 + ch) * ovol + o3] = val;
    else        img_out[(long long)b * ovol + o3] = val;
  }
}

// ---------------------------------------------------------------------------
// Deterministic final reduction: sum(psum_region_i * invN_i)
// ---------------------------------------------------------------------------
__global__ __launch_bounds__(256) void final_reduce_kernel(
    const float* __restrict__ psum, int n0, int n1, int n2,
    float w0, float w1, float w2, float* __restrict__ out)
{
  __shared__ float red[256];
  int tid = threadIdx.x;
  int total = n0 + n1 + n2;
  float acc = 0.0f;
  for (int i = tid; i < total; i += 256) {
    float w = (i < n0) ? w0 : ((i < n0 + n1) ? w1 : w2);
    acc += psum[i] * w;
  }
  red[tid] = acc;
  __syncthreads();
  for (int s = 128; s > 0; s >>= 1) {
    if (tid < s) red[tid] += red[tid + s];
    __syncthreads();
  }
  if (tid == 0) out[0] = red[0];
}

// ---------------------------------------------------------------------------
static inline int ceil_div(int a, int b) { return (a + b - 1) / b; }

extern "C" void kernel_launch(void* const* d_in, const int* in_sizes, int n_in,
                              void* d_out, int out_size, void* d_ws, size_t ws_size,
                              hipStream_t stream) {
  (void)in_sizes; (void)n_in; (void)out_size; (void)ws_size;
  const float* def = (const float*)d_in[0];   // (2,3,160,192,160)
  const float* img = (const float*)d_in[1];   // (2,1,160,192,160)
  float* out = (float*)d_out;
  float* ws = (float*)d_ws;

  const int B = 2;
  const int D0 = 160, D1 = 192, D2 = 160;   // scale 0
  const int A0 = 80,  A1 = 96,  A2 = 80;    // scale 1 (floor /2)
  const int C0 = 40,  C1 = 48,  C2 = 40;    // scale 2 (floor /4)
  const long long vol0 = (long long)D0 * D1 * D2;
  const long long vol1 = (long long)A0 * A1 * A2;
  const long long vol2 = (long long)C0 * C1 * C2;

  // workspace layout
  float* def1 = ws;                               // B*3*vol1
  float* img1 = def1 + (long long)B * 3 * vol1;   // B*vol1
  float* def2 = img1 + (long long)B * vol1;       // B*3*vol2
  float* img2 = def2 + (long long)B * 3 * vol2;   // B*vol2
  float* psum = img2 + (long long)B * vol2;

  // tile counts per scale
  const int tx0 = ceil_div(D0, TX), ty0 = ceil_div(D1, TY), tz0 = ceil_div(D2, TZ);
  const int tx1 = ceil_div(A0, TX), ty1 = ceil_div(A1, TY), tz1 = ceil_div(A2, TZ);
  const int tx2 = ceil_div(C0, TX), ty2 = ceil_div(C1, TY), tz2 = ceil_div(C2, TZ);
  const int nt0 = tx0 * ty0 * tz0;   // 2400
  const int nt1 = tx1 * ty1 * tz1;   // 360
  const int nt2 = tx2 * ty2 * tz2;   // 60

  float* ps0 = psum;
  float* ps1 = ps0 + (long long)B * nt0;
  float* ps2 = ps1 + (long long)B * nt1;

  // --- downsample scale 1 and 2 from the original volumes ---
  {
    long long total = (long long)B * NCH * vol1;
    int blocks = (int)((total + 255) / 256);
    downsample_kernel<<<blocks, 256, 0, stream>>>(
        def, img, D0, D1, D2, A0, A1, A2,
        (float)((double)(D0 - 1) / (A0 - 1)),
        (float)((double)(D1 - 1) / (A1 - 1)),
        (float)((double)(D2 - 1) / (A2 - 1)),
        def1, img1);
  }
  {
    long long total = (long long)B * NCH * vol2;
    int blocks = (int)((total + 255) / 256);
    downsample_kernel<<<blocks, 256, 0, stream>>>(
        def, img, D0, D1, D2, C0, C1, C2,
        (float)((double)(D0 - 1) / (C0 - 1)),
        (float)((double)(D1 - 1) / (C1 - 1)),
        (float)((double)(D2 - 1) / (C2 - 1)),
        def2, img2);
  }

  // --- stencil per scale (partial sums, deterministic) ---
  elasticity_tile_kernel<<<dim3(nt0, B), 256, 0, stream>>>(
      def, img, D0, D1, D2, ty0, tz0, nt0, ps0);
  elasticity_tile_kernel<<<dim3(nt1, B), 256, 0, stream>>>(
      def1, img1, A0, A1, A2, ty1, tz1, nt1, ps1);
  elasticity_tile_kernel<<<dim3(nt2, B), 256, 0, stream>>>(
      def2, img2, C0, C1, C2, ty2, tz2, nt2, ps2);

  // --- final deterministic reduction with per-scale 1/N weights ---
  final_reduce_kernel<<<1, 256, 0, stream>>>(
      psum, B * nt0, B * nt1, B * nt2,
      (float)(1.0 / ((double)B * vol0)),
      (float)(1.0 / ((double)B * vol1)),
      (float)(1.0 / ((double)B * vol2)),
      out);
}